// AKF_66881230733821
// MI455X (gfx1250) — compile-verified
//
#include <hip/hip_runtime.h>

typedef __attribute__((ext_vector_type(2))) float v2f;
typedef __attribute__((ext_vector_type(8))) float v8f;

#define SDIM 64
#define MDIM 32

// ---------------------------------------------------------------------------
// Preamble: compute K (64x32) and c (64) into workspace.
//   x_pred = F x
//   P_pred = F P F^T + Q
//   S      = H P_pred H^T + R
//   K      = P_pred H^T inv(S)
//   c      = x_pred - K (H x_pred)
// Single block, 256 threads, staged in 48KB of LDS.
// ---------------------------------------------------------------------------
__global__ __launch_bounds__(256) void akf_prep(
    const float* __restrict__ F, const float* __restrict__ H,
    const float* __restrict__ Q, const float* __restrict__ R,
    const float* __restrict__ P, const float* __restrict__ x,
    float* __restrict__ Kout, float* __restrict__ cout)
{
    __shared__ float sA[SDIM * SDIM];   // F; later reused: [0..2047]=H, [2048..]=S_aug
    __shared__ float sB[SDIM * SDIM];   // P -> P_pred
    __shared__ float sC[SDIM * SDIM];   // T1 -> T2 -> T3 (low half), K (high half)
    __shared__ float xp[SDIM];
    __shared__ float hx[MDIM];
    __shared__ float scol[MDIM];
    __shared__ float piv;

    const int t = threadIdx.x;

    for (int i = t; i < SDIM * SDIM; i += 256) { sA[i] = F[i]; sB[i] = P[i]; }
    __syncthreads();

    if (t < SDIM) {
        float s = 0.f;
        for (int k = 0; k < SDIM; k++) s += sA[t * SDIM + k] * x[k];
        xp[t] = s;
    }
    for (int i = t; i < SDIM * SDIM; i += 256) {
        int r = i >> 6, c = i & 63;
        float s = 0.f;
        for (int k = 0; k < SDIM; k++) s += sA[r * SDIM + k] * sB[k * SDIM + c];
        sC[i] = s;
    }
    __syncthreads();

    for (int i = t; i < SDIM * SDIM; i += 256) {
        int r = i >> 6, c = i & 63;
        float s = Q[i];
        for (int k = 0; k < SDIM; k++) s += sC[r * SDIM + k] * sA[c * SDIM + k];
        sB[i] = s;
    }
    __syncthreads();

    float* sH  = sA;
    float* sSa = sA + 2048;
    for (int i = t; i < MDIM * SDIM; i += 256) sH[i] = H[i];
    __syncthreads();

    if (t < MDIM) {
        float s = 0.f;
        for (int j = 0; j < SDIM; j++) s += sH[t * SDIM + j] * xp[j];
        hx[t] = s;
    }
    for (int i = t; i < MDIM * SDIM; i += 256) {
        int r = i >> 6, c = i & 63;
        float s = 0.f;
        for (int k = 0; k < SDIM; k++) s += sH[r * SDIM + k] * sB[k * SDIM + c];
        sC[i] = s;
    }
    __syncthreads();

    for (int i = t; i < MDIM * 64; i += 256) {
        int r = i >> 6, c = i & 63;
        float v;
        if (c < MDIM) {
            float s = R[r * MDIM + c];
            for (int k = 0; k < SDIM; k++) s += sC[r * SDIM + k] * sH[c * SDIM + k];
            v = s;
        } else {
            v = ((c - MDIM) == r) ? 1.f : 0.f;
        }
        sSa[i] = v;
    }
    __syncthreads();

    for (int p = 0; p < MDIM; p++) {
        if (t == 0) piv = 1.0f / sSa[p * 64 + p];
        __syncthreads();
        if (t < 64) sSa[p * 64 + t] *= piv;
        else if (t < 96) { int r = t - 64; scol[r] = sSa[r * 64 + p]; }
        __syncthreads();
        for (int i = t; i < MDIM * 64; i += 256) {
            int r = i >> 6, c = i & 63;
            if (r != p) sSa[i] -= scol[r] * sSa[p * 64 + c];
        }
        __syncthreads();
    }

    for (int i = t; i < SDIM * MDIM; i += 256) {
        int r = i >> 5, c = i & 31;
        float s = 0.f;
        for (int k = 0; k < SDIM; k++) s += sB[r * SDIM + k] * sH[c * SDIM + k];
        sC[i] = s;
    }
    __syncthreads();

    for (int i = t; i < SDIM * MDIM; i += 256) {
        int r = i >> 5, c = i & 31;
        float s = 0.f;
        for (int k = 0; k < MDIM; k++) s += sC[r * MDIM + k] * sSa[k * 64 + MDIM + c];
        sC[2048 + i] = s;
    }
    __syncthreads();

    for (int i = t; i < SDIM * MDIM; i += 256) Kout[i] = sC[2048 + i];
    if (t < SDIM) {
        float s = xp[t];
        for (int j = 0; j < MDIM; j++) s -= sC[2048 + t * MDIM + j] * hx[j];
        cout[t] = s;
    }
}

// ---------------------------------------------------------------------------
// Main GEMM: out[64, N] = c + K(64x32) @ z(32, N).
// One wave per 64x16 tile, 8 tiles per wave. All K fragments and c hoisted
// into registers once; steady-state per tile = 16 NT b32 loads (one clause),
// 32 chained v_wmma_f32_16x16x4_f32, 32 NT b32 stores. 32-bit offset math.
// ---------------------------------------------------------------------------
#define KROW 33
#define COFF (SDIM * KROW)
#define TPW  8                      // tiles per wave
#define COLS_PER_BLOCK (8 * TPW * 16)

__global__ __launch_bounds__(256) void akf_apply(
    const float* __restrict__ z, const float* __restrict__ KC,
    float* __restrict__ out, unsigned N)
{
    __shared__ float sK[COFF + SDIM];

    for (int i = threadIdx.x; i < SDIM * MDIM; i += 256)
        sK[(i >> 5) * KROW + (i & 31)] = KC[i];
    for (int i = threadIdx.x; i < SDIM; i += 256)
        sK[COFF + i] = KC[SDIM * MDIM + i];
    __syncthreads();

    const int wave = threadIdx.x >> 5;
    const int lane = threadIdx.x & 31;
    const int half = lane >> 4;      // 0: lanes 0-15, 1: lanes 16-31
    const int ln   = lane & 15;

    // Hoist all A fragments (16x4 f32 ISA layout): lanes 0-15 carry K=k0,k0+1;
    // lanes 16-31 carry K=k0+2,k0+3 of row M=ln (+16*mt).
    v2f a[8][4];
#pragma unroll
    for (int kk = 0; kk < 8; kk++) {
        const int k0 = kk * 4 + 2 * half;
#pragma unroll
        for (int mt = 0; mt < 4; mt++) {
            a[kk][mt].x = sK[(mt * 16 + ln) * KROW + k0];
            a[kk][mt].y = sK[(mt * 16 + ln) * KROW + k0 + 1];
        }
    }
    // Fold c into the accumulator init (16x16 f32 C/D layout).
    v8f cinit[4];
#pragma unroll
    for (int mt = 0; mt < 4; mt++)
#pragma unroll
        for (int r = 0; r < 8; r++)
            cinit[mt][r] = sK[COFF + mt * 16 + half * 8 + r];

    unsigned col = ((unsigned)blockIdx.x * 8 + (unsigned)wave) * (TPW * 16)
                   + (unsigned)ln;

    const float* zb = z   + (unsigned)(2 * half) * N;   // lane-half K-row base
    float*       ob = out + (unsigned)(half * 8) * N;   // lane-half M-row base

#pragma unroll 1
    for (int tt = 0; tt < TPW; tt++, col += 16) {
        // Batch all 16 z loads for this tile into one clause.
        v2f b[8];
#pragma unroll
        for (int kk = 0; kk < 8; kk++) {
            b[kk].x = __builtin_nontemporal_load(zb + (unsigned)(kk * 4 + 0) * N + col);
            b[kk].y = __builtin_nontemporal_load(zb + (unsigned)(kk * 4 + 1) * N + col);
        }

        v8f acc[4] = { cinit[0], cinit[1], cinit[2], cinit[3] };
#pragma unroll
        for (int kk = 0; kk < 8; kk++)
#pragma unroll
            for (int mt = 0; mt < 4; mt++)
                acc[mt] = __builtin_amdgcn_wmma_f32_16x16x4_f32(
                    false, a[kk][mt], false, b[kk],
                    (short)0, acc[mt], false, false);

#pragma unroll
        for (int mt = 0; mt < 4; mt++)
#pragma unroll
            for (int r = 0; r < 8; r++)
                __builtin_nontemporal_store(acc[mt][r],
                    ob + (unsigned)(mt * 16 + r) * N + col);
    }
}

// Scalar tail for N not a multiple of COLS_PER_BLOCK (not hit for N = 2^20).
__global__ __launch_bounds__(256) void akf_tail(
    const float* __restrict__ z, const float* __restrict__ KC,
    float* __restrict__ out, unsigned N, unsigned start)
{
    unsigned j = start + blockIdx.x * blockDim.x + threadIdx.x;
    if (j >= N) return;
    float zc[MDIM];
#pragma unroll
    for (int k = 0; k < MDIM; k++) zc[k] = z[(unsigned)k * N + j];
#pragma unroll 4
    for (int r = 0; r < SDIM; r++) {
        float s = KC[SDIM * MDIM + r];
#pragma unroll
        for (int k = 0; k < MDIM; k++) s += KC[r * MDIM + k] * zc[k];
        out[(unsigned)r * N + j] = s;
    }
}

extern "C" void kernel_launch(void* const* d_in, const int* in_sizes, int n_in,
                              void* d_out, int out_size, void* d_ws, size_t ws_size,
                              hipStream_t stream) {
    const float* z = (const float*)d_in[0];
    const float* F = (const float*)d_in[1];
    const float* H = (const float*)d_in[2];
    const float* Q = (const float*)d_in[3];
    const float* R = (const float*)d_in[4];
    const float* P = (const float*)d_in[5];
    const float* x = (const float*)d_in[6];
    float* out = (float*)d_out;
    float* ws  = (float*)d_ws;            // [0..2047]=K, [2048..2111]=c

    const unsigned N = (unsigned)(in_sizes[0] / MDIM);

    akf_prep<<<1, 256, 0, stream>>>(F, H, Q, R, P, x, ws, ws + SDIM * MDIM);

    const unsigned nmain = N & ~(unsigned)(COLS_PER_BLOCK - 1);
    if (nmain > 0)
        akf_apply<<<(int)(nmain / COLS_PER_BLOCK), 256, 0, stream>>>(z, ws, out, N);
    const unsigned rem = N - nmain;
    if (rem > 0)
        akf_tail<<<(int)((rem + 255) / 256), 256, 0, stream>>>(z, ws, out, N, nmain);
}